// GAT_37185826849273
// MI455X (gfx1250) — compile-verified
//
#include <hip/hip_runtime.h>

#define NEG_SLOPE 0.2f
#define ENC_NEG_INF 0x007FFFFFu   // monotone encoding of -inf

typedef float v2f __attribute__((ext_vector_type(2)));
typedef float v8f __attribute__((ext_vector_type(8)));

// ---- monotone float <-> u32 map so segment-max can use atomicMax(u32) ----
__device__ __forceinline__ unsigned gat_fenc(float f) {
    unsigned u = __float_as_uint(f);
    return (u & 0x80000000u) ? ~u : (u | 0x80000000u);
}
__device__ __forceinline__ float gat_fdec(unsigned u) {
    unsigned b = (u & 0x80000000u) ? (u & 0x7FFFFFFFu) : ~u;
    return __uint_as_float(b);
}

// ---------------------------------------------------------------------------
// GEMM  H[n_rows x (NT*16)] = X[n_rows x 128] * W[128 x (NT*16)]
// via V_WMMA_F32_16X16X4_F32. One wave = 16 rows x full width.
// W is staged transposed into LDS with additive swizzle (k+4n)&127 so the
// per-half-wave ds_load_b64 of the B k-pair is bank-conflict-free.
// ---------------------------------------------------------------------------
template<int NT>
__global__ __launch_bounds__(256)
void gat_gemm_wmma(const float* __restrict__ X, const float* __restrict__ W,
                   float* __restrict__ Hout, int n_rows)
{
    constexpr int KD = 128;
    constexpr int ND = NT * 16;
    __shared__ float Wt[ND * KD];          // 64 KB for NT=8, 16 KB for NT=2

    for (int i = threadIdx.x; i < KD * ND; i += 256) {
        int k = i / ND, n = i % ND;        // consecutive threads -> coalesced W read
        Wt[n * KD + ((k + 4 * n) & (KD - 1))] = W[i];
    }
    __syncthreads();

    const int lane = threadIdx.x & 31;
    const int wave = threadIdx.x >> 5;
    const int m0   = (blockIdx.x * 8 + wave) * 16;
    const int mlo  = lane & 15;            // A: M index / B: N index
    const int kofs = (lane >> 4) * 2;      // half-wave K offset (f32 WMMA layout)

    int arow = m0 + mlo;
    if (arow >= n_rows) arow = n_rows - 1; // clamp: keep EXEC all-ones for WMMA
    const float* xrow = X + (size_t)arow * KD + kofs;

    v8f acc[NT] = {};

    for (int k = 0; k < KD; k += 4) {
        v2f a = *(const v2f*)(xrow + k);   // A vgpr0/1 = K(kofs), K(kofs+1)
#pragma unroll
        for (int t = 0; t < NT; ++t) {
            int n = t * 16 + mlo;
            v2f b = *(const v2f*)(&Wt[n * KD + ((k + kofs + 4 * n) & (KD - 1))]);
            acc[t] = __builtin_amdgcn_wmma_f32_16x16x4_f32(
                false, a, false, b, (short)0, acc[t], false, false);
        }
    }

    const int mhi = (lane >> 4) * 8;       // C/D layout: VGPR j -> M=j(+8)
#pragma unroll
    for (int t = 0; t < NT; ++t)
#pragma unroll
        for (int j = 0; j < 8; ++j) {
            int m = m0 + mhi + j;
            if (m < n_rows) Hout[(size_t)m * ND + t * 16 + mlo] = acc[t][j];
        }
}

// ---------------------------------------------------------------------------
// Per-node attention scores: a_src[n,h] = <H[n,h,:], att_src[h,:]>, same dst
// ---------------------------------------------------------------------------
__global__ void gat_attn_score(const float* __restrict__ Hf,
                               const float* __restrict__ att_s,
                               const float* __restrict__ att_d,
                               float* __restrict__ a_s, float* __restrict__ a_d,
                               int n, int heads, int ch)
{
    int t = blockIdx.x * blockDim.x + threadIdx.x;
    if (t >= n * heads) return;
    int node = t / heads, h = t - node * heads;
    const float* hp = Hf + ((size_t)node * heads + h) * ch;
    const float* sp = att_s + h * ch;
    const float* dp = att_d + h * ch;
    float ss = 0.f, dd = 0.f;
    for (int c = 0; c < ch; ++c) { float v = hp[c]; ss += v * sp[c]; dd += v * dp[c]; }
    a_s[t] = ss; a_d[t] = dd;
}

// ---------------------------------------------------------------------------
__global__ void gat_fill_u32(unsigned* __restrict__ p, unsigned v, int n)
{
    int i = blockIdx.x * blockDim.x + threadIdx.x;
    if (i < n) p[i] = v;
}

__global__ void gat_bias_init(float* __restrict__ out, const float* __restrict__ bias,
                              int total, int dim)
{
    int i = blockIdx.x * blockDim.x + threadIdx.x;
    if (i < total) out[i] = bias[i % dim];
}

__global__ void gat_relu(float* __restrict__ p, int n)
{
    int i = blockIdx.x * blockDim.x + threadIdx.x;
    if (i < n) p[i] = fmaxf(p[i], 0.f);
}

// ---------------------------------------------------------------------------
// Edge pass 1: e = leakyrelu(a_src[src]+a_dst[dst]); segment-max into emax[dst]
// ---------------------------------------------------------------------------
__global__ void gat_edge_scores(const int* __restrict__ srcI, const int* __restrict__ dstI,
                                const float* __restrict__ a_s, const float* __restrict__ a_d,
                                float* __restrict__ eb, unsigned* __restrict__ emax,
                                int E, int E2, int H)
{
    int e = blockIdx.x * blockDim.x + threadIdx.x;
    if (e >= E2) return;
    int s = (e < E) ? srcI[e] : (e - E);   // self-loops appended after real edges
    int d = (e < E) ? dstI[e] : (e - E);
    for (int h = 0; h < H; ++h) {
        float v = a_s[s * H + h] + a_d[d * H + h];
        v = (v >= 0.f) ? v : NEG_SLOPE * v;
        eb[(size_t)e * H + h] = v;
        atomicMax(&emax[d * H + h], gat_fenc(v));
    }
}

// Edge pass 2: eb = exp(e - emax[dst]); segment-sum into denom[dst]
__global__ void gat_edge_exp(const int* __restrict__ dstI,
                             float* __restrict__ eb,
                             const unsigned* __restrict__ emax,
                             float* __restrict__ denom,
                             int E, int E2, int H)
{
    int e = blockIdx.x * blockDim.x + threadIdx.x;
    if (e >= E2) return;
    int d = (e < E) ? dstI[e] : (e - E);
    for (int h = 0; h < H; ++h) {
        float m = gat_fdec(emax[d * H + h]);
        float x = __expf(eb[(size_t)e * H + h] - m);
        eb[(size_t)e * H + h] = x;
        atomicAdd(&denom[d * H + h], x);
    }
}

// Edge pass 3: out[dst,h,:] += (eb/denom[dst,h]) * H[src,h,:]   (C = 32 fixed)
__global__ void gat_edge_aggregate(const int* __restrict__ srcI, const int* __restrict__ dstI,
                                   const float* __restrict__ eb, const float* __restrict__ denom,
                                   const float* __restrict__ Hf, float* __restrict__ outp,
                                   int E, int E2, int H)
{
    int t = blockIdx.x * blockDim.x + threadIdx.x;
    if (t >= E2 * H) return;
    int e = t / H, h = t - e * H;
    int s = (e < E) ? srcI[e] : (e - E);
    int d = (e < E) ? dstI[e] : (e - E);
    float alpha = eb[(size_t)e * H + h] / denom[d * H + h];
    const float4* hv = (const float4*)(Hf + ((size_t)s * H + h) * 32);
    float*       ov = outp + ((size_t)d * H + h) * 32;
#pragma unroll
    for (int i = 0; i < 8; ++i) {
        float4 v = hv[i];
        atomicAdd(ov + 4 * i + 0, alpha * v.x);
        atomicAdd(ov + 4 * i + 1, alpha * v.y);
        atomicAdd(ov + 4 * i + 2, alpha * v.z);
        atomicAdd(ov + 4 * i + 3, alpha * v.w);
    }
}

// ---------------------------------------------------------------------------
extern "C" void kernel_launch(void* const* d_in, const int* in_sizes, int n_in,
                              void* d_out, int out_size, void* d_ws, size_t ws_size,
                              hipStream_t stream)
{
    const float* x   = (const float*)d_in[0];
    const int*   ei  = (const int*)d_in[1];
    const float* W1  = (const float*)d_in[2];
    const float* as1 = (const float*)d_in[3];
    const float* ad1 = (const float*)d_in[4];
    const float* b1  = (const float*)d_in[5];
    const float* W2  = (const float*)d_in[6];
    const float* as2 = (const float*)d_in[7];
    const float* ad2 = (const float*)d_in[8];
    const float* b2  = (const float*)d_in[9];
    float* out = (float*)d_out;

    const int N  = in_sizes[0] / 128;
    const int E  = in_sizes[1] / 2;
    const int E2 = E + N;                        // + self-loops
    const int* srcI = ei;                        // edge_index[0]
    const int* dstI = ei + E;                    // edge_index[1]

    // ---- workspace carve (~79 MB) ----
    float* ws = (float*)d_ws;
    float*    h1    = ws;              ws += (size_t)N * 128;
    float*    asrc1 = ws;              ws += (size_t)N * 4;
    float*    adst1 = ws;              ws += (size_t)N * 4;
    unsigned* emax1 = (unsigned*)ws;   ws += (size_t)N * 4;
    float*    den1  = ws;              ws += (size_t)N * 4;
    float*    eb1   = ws;              ws += (size_t)E2 * 4;
    float*    h2    = ws;              ws += (size_t)N * 128;  // out1, relu'd in place
    float*    g2    = ws;              ws += (size_t)N * 32;
    float*    asrc2 = ws;              ws += (size_t)N;
    float*    adst2 = ws;              ws += (size_t)N;
    unsigned* emax2 = (unsigned*)ws;   ws += (size_t)N;
    float*    den2  = ws;              ws += (size_t)E2 > 0 ? (size_t)N : 0;
    float*    eb2   = ws;

    const int TB = 256;
    const int gB   = (N + 127) / 128;            // gemm blocks (8 waves * 16 rows)
    const int gE   = (E2 + TB - 1) / TB;
    const int gE4  = (E2 * 4 + TB - 1) / TB;
    const int gN4  = (N * 4 + TB - 1) / TB;
    const int gN1  = (N + TB - 1) / TB;
    const int gNF  = (N * 128 + TB - 1) / TB;
    const int gNO  = (N * 32 + TB - 1) / TB;

    // ================= Layer 1 (H=4, C=32, concat) =================
    gat_gemm_wmma<8><<<gB, TB, 0, stream>>>(x, W1, h1, N);
    gat_attn_score<<<gN4, TB, 0, stream>>>(h1, as1, ad1, asrc1, adst1, N, 4, 32);
    gat_fill_u32<<<gN4, TB, 0, stream>>>(emax1, ENC_NEG_INF, N * 4);
    gat_fill_u32<<<gN4, TB, 0, stream>>>((unsigned*)den1, 0u, N * 4);
    gat_bias_init<<<gNF, TB, 0, stream>>>(h2, b1, N * 128, 128);
    gat_edge_scores<<<gE, TB, 0, stream>>>(srcI, dstI, asrc1, adst1, eb1, emax1, E, E2, 4);
    gat_edge_exp<<<gE, TB, 0, stream>>>(dstI, eb1, emax1, den1, E, E2, 4);
    gat_edge_aggregate<<<gE4, TB, 0, stream>>>(srcI, dstI, eb1, den1, h1, h2, E, E2, 4);
    gat_relu<<<gNF, TB, 0, stream>>>(h2, N * 128);

    // ================= Layer 2 (H=1, C=32, mean over 1 head) =================
    gat_gemm_wmma<2><<<gB, TB, 0, stream>>>(h2, W2, g2, N);
    gat_attn_score<<<gN1, TB, 0, stream>>>(g2, as2, ad2, asrc2, adst2, N, 1, 32);
    gat_fill_u32<<<gN1, TB, 0, stream>>>(emax2, ENC_NEG_INF, N);
    gat_fill_u32<<<gN1, TB, 0, stream>>>((unsigned*)den2, 0u, N);
    gat_bias_init<<<gNO, TB, 0, stream>>>(out, b2, N * 32, 32);
    gat_edge_scores<<<gE, TB, 0, stream>>>(srcI, dstI, asrc2, adst2, eb2, emax2, E, E2, 1);
    gat_edge_exp<<<gE, TB, 0, stream>>>(dstI, eb2, emax2, den2, E, E2, 1);
    gat_edge_aggregate<<<gE, TB, 0, stream>>>(srcI, dstI, eb2, den2, g2, out, E, E2, 1);
}